// NCNLinearKernel_57724360458413
// MI455X (gfx1250) — compile-verified
//
#include <hip/hip_runtime.h>
#include <math.h>

typedef __attribute__((ext_vector_type(2))) float v2f;
typedef __attribute__((ext_vector_type(8))) float v8f;

#define NB     16
#define NL     2048
#define ND     128
#define NGROUP 16
#define NNG    (NL / NGROUP)   // 128
#define NALPHA 0.5f
#define DPAD   (ND + 4)        // LDS row pad: stride 132 dwords -> lanes land on distinct banks

// Fast tanh: prefer the gfx1250 hardware v_tanh_f32; otherwise 2 TRANS ops
// (v_exp_f32 + v_rcp_f32):  tanh(x) = 1 - 2 / (exp2(2*log2e*x) + 1)
// Saturates correctly: x>>0 -> exp2=inf -> rcp=0 -> +1 ; x<<0 -> exp2=0 -> -1.
__device__ __forceinline__ float fast_tanh(float x) {
#if __has_builtin(__builtin_amdgcn_tanhf)
    return __builtin_amdgcn_tanhf(x);
#else
    const float e = __builtin_amdgcn_exp2f(x * 2.885390081777927f); // 2*log2(e)
    return 1.0f - 2.0f * __builtin_amdgcn_rcpf(e + 1.0f);
#endif
}

__launch_bounds__(256, 2)
__global__ void ncn_forward_kernel(const float* __restrict__ x,
                                   const int*   __restrict__ group_tensor,
                                   const int*   __restrict__ ctxlens,
                                   const float* __restrict__ W,
                                   const float* __restrict__ norm_alpha,
                                   const float* __restrict__ norm_gamma,
                                   const float* __restrict__ norm_beta,
                                   float* __restrict__ yi_out,
                                   float* __restrict__ ya_out)
{
    __shared__ float s_xi[NGROUP][DPAD];
    __shared__ float s_xa[NGROUP][DPAD];
    __shared__ float s_dots[2][NGROUP];   // [0]=xi·Wi per row, [1]=xi·Wj per row
    __shared__ int   s_idx[NGROUP];
    __shared__ float s_gamma[2 * ND];
    __shared__ float s_beta[2 * ND];

    const int blk  = blockIdx.x;          // 0 .. NB*NNG-1
    const int b    = blk / NNG;
    const int g    = blk % NNG;
    const int tid  = threadIdx.x;         // 0..255
    const int lane = tid & 31;
    const int wave = tid >> 5;

    // Stage norm params + permutation indices for this group into LDS.
    for (int i = tid; i < 2 * ND; i += 256) {
        s_gamma[i] = norm_gamma[i];
        s_beta[i]  = norm_beta[i];
    }
    if (tid < NGROUP) s_idx[tid] = group_tensor[b * NL + g * NGROUP + tid];
    __syncthreads();

    const float a1 = norm_alpha[0];
    const float a2 = norm_alpha[1];

    // Gather: thread t owns row r = t>>4, dims d0..d0+7 with d0 = (t&15)*8.
    const int r  = tid >> 4;
    const int d0 = (tid & 15) * 8;
    {
        const float* src = x + ((size_t)b * NL + (size_t)s_idx[r]) * ND + d0;
        #pragma unroll
        for (int i = 0; i < 8; ++i) {
            s_xi[r][d0 + i] = src[i];
            s_xa[r][d0 + i] = 0.0f;
        }
    }

    // Wave 0: build the constant B fragments once.
    // B (4x16, KxN) per chunk c: col 0 = Wi[4c..4c+3], col 1 = Wj[4c..4c+3], cols 2..15 = 0.
    // Lane layout: n = lane&15, half = lane>>4; VGPR0 = B[2*half][n], VGPR1 = B[2*half+1][n].
    v2f bfrag[32];
    const int half = (lane >> 4) & 1;
    const int mrow = lane & 15;
    if (wave == 0) {
        #pragma unroll
        for (int c = 0; c < 32; ++c) {
            float b0 = 0.0f, b1 = 0.0f;
            if (mrow < 2) {
                const float* Wp = W + mrow * ND;   // row 0 -> Wi, row 1 -> Wj
                b0 = Wp[4 * c + 2 * half];
                b1 = Wp[4 * c + 2 * half + 1];
            }
            bfrag[c].x = b0;
            bfrag[c].y = b1;
        }
    }
    __syncthreads();

    for (int j = 0; j < NGROUP; ++j) {
        // ---- Dot products via v_wmma_f32_16x16x4_f32 (wave 0, EXEC all-ones) ----
        if (wave == 0) {
            v8f acc = {};
            #pragma unroll
            for (int c = 0; c < 32; ++c) {
                // A (16x4): lane m holds row m, K = 2*half + {0,1} of chunk c.
                v2f a;
                a.x = s_xi[mrow][4 * c + 2 * half];
                a.y = s_xi[mrow][4 * c + 2 * half + 1];
                acc = __builtin_amdgcn_wmma_f32_16x16x4_f32(
                    /*neg_a=*/false, a, /*neg_b=*/false, bfrag[c],
                    /*c_mod=*/(short)0, acc, /*reuse_a=*/false, /*reuse_b=*/false);
            }
            // D layout: lanes 0..15 -> N=lane, rows 0..7 in acc[0..7]; lanes 16..31 -> rows 8..15.
            // Column 0 = xi·Wi dots, column 1 = xi·Wj dots.
            if (mrow < 2) {
                const int rbase = half * 8;
                #pragma unroll
                for (int v = 0; v < 8; ++v)
                    s_dots[mrow][rbase + v] = acc[v];
            }
        }
        __syncthreads();

        const float simj  = s_dots[1][j];          // xj · Wj
        const float sim_r = s_dots[0][r] + simj;   // sim for this thread's row

        // Snapshot pre-update values (row j is read by everyone, written by its owners).
        float xi_v[8], xj_v[8], xa_v[8];
        #pragma unroll
        for (int i = 0; i < 8; ++i) {
            xi_v[i] = s_xi[r][d0 + i];
            xj_v[i] = s_xi[j][d0 + i];
            xa_v[i] = s_xa[r][d0 + i];
        }
        __syncthreads();

        #pragma unroll
        for (int i = 0; i < 8; ++i) {
            const int d  = d0 + i;
            const float T  = NALPHA * xi_v[i] + (1.0f - NALPHA) * sim_r * xj_v[i];
            const float Tn = s_gamma[d] * fast_tanh(a1 * T) + s_beta[d];
            const float Fv = (Tn >= 0.0f) ? Tn : 0.01f * Tn;
            const float xa_new = xa_v[i] + Fv;
            const float xi_new = xi_v[i] + s_gamma[ND + d] * fast_tanh(a2 * xa_new) + s_beta[ND + d];
            s_xa[r][d] = xa_new;
            s_xi[r][d] = xi_new;
        }
        __syncthreads();
    }

    // Scatter results back through the permutation (mask by ctxlens).
    const bool valid = (g * NGROUP) < ctxlens[b];
    float* yi_p = yi_out + ((size_t)b * NL + (size_t)s_idx[r]) * ND + d0;
    float* ya_p = ya_out + ((size_t)b * NL + (size_t)s_idx[r]) * ND + d0;
    #pragma unroll
    for (int i = 0; i < 8; ++i) {
        yi_p[i] = valid ? s_xi[r][d0 + i] : 0.0f;
        ya_p[i] = valid ? s_xa[r][d0 + i] : 0.0f;
    }
}

extern "C" void kernel_launch(void* const* d_in, const int* in_sizes, int n_in,
                              void* d_out, int out_size, void* d_ws, size_t ws_size,
                              hipStream_t stream) {
    const float* x            = (const float*)d_in[0];
    const int*   group_tensor = (const int*)d_in[1];
    const int*   ctxlens      = (const int*)d_in[2];
    const float* W            = (const float*)d_in[3];
    const float* norm_alpha   = (const float*)d_in[4];
    const float* norm_gamma   = (const float*)d_in[5];
    const float* norm_beta    = (const float*)d_in[6];

    float* yi_out = (float*)d_out;
    float* ya_out = yi_out + (size_t)NB * NL * ND;

    dim3 grid(NB * NNG);   // 2048 groups, one per block
    dim3 block(256);       // 8 wave32
    ncn_forward_kernel<<<grid, block, 0, stream>>>(
        x, group_tensor, ctxlens, W, norm_alpha, norm_gamma, norm_beta, yi_out, ya_out);
}